// Capsule_49658411876931
// MI455X (gfx1250) — compile-verified
//
#include <hip/hip_runtime.h>

typedef __attribute__((ext_vector_type(2))) float v2f;
typedef __attribute__((ext_vector_type(8))) float v8f;

#define B_ 32
#define I_ 2048
#define K_ 8
#define J_ 32
#define D_ 16
#define CH 64   // i-chunk per routing block

// ---------------------------------------------------------------------------
// Kernel 1: u_hat[j][i][b][d] = sum_k W[j,i,d,k] * x[b,i,k]  via f32 WMMA.
// One block per i (2048 blocks, 8 waves). Per wave: 4 j-tiles x 2 b-halves,
// each accumulated with two v_wmma_f32_16x16x4_f32 (K=8 = 2 k-steps).
// ---------------------------------------------------------------------------
__global__ __launch_bounds__(256) void uhat_wmma_kernel(
    const float* __restrict__ x, const float* __restrict__ W,
    float* __restrict__ uhat) {
  const int i    = blockIdx.x;
  const int t    = threadIdx.x;
  const int wave = t >> 5;
  const int lane = t & 31;

  __shared__ float xs[B_ * K_];              // x[b][k] slice for this i (1 KB)
  if (t < B_ * K_) {
    const int b = t >> 3, k = t & 7;
    xs[t] = x[(size_t)(b * I_ + i) * K_ + k];
  }
  __syncthreads();

  const int n  = lane & 15;                  // N index (b within half)
  const int hi = lane >> 4;                  // lane-group: K offset selector

  // B fragments (4x16, f32): lane L holds K = ks*4 + 2*hi + {0,1}, col n.
  v2f bfrag[2][2];
  #pragma unroll
  for (int h = 0; h < 2; ++h) {
    const float* xr = &xs[(h * 16 + n) * K_];
    #pragma unroll
    for (int ks = 0; ks < 2; ++ks)
      bfrag[h][ks] = *(const v2f*)(xr + ks * 4 + 2 * hi);
  }

  const int m = n;                           // A row = d index
  #pragma unroll
  for (int jt = 0; jt < 4; ++jt) {
    const int j = jt * 8 + wave;
    const size_t wbase = ((size_t)(j * I_ + i) * D_ + m) * K_;
    const v2f a0 = *(const v2f*)(W + wbase + 2 * hi);        // k = 0..3 group
    const v2f a1 = *(const v2f*)(W + wbase + 4 + 2 * hi);    // k = 4..7 group

    #pragma unroll
    for (int h = 0; h < 2; ++h) {
      v8f c = {};
      c = __builtin_amdgcn_wmma_f32_16x16x4_f32(false, a0, false, bfrag[h][0],
                                                (short)0, c, false, false);
      c = __builtin_amdgcn_wmma_f32_16x16x4_f32(false, a1, false, bfrag[h][1],
                                                (short)0, c, false, false);
      // D layout: lane L, VGPR v -> M = v + 8*(L/16), N = L%16.
      // u_hat[j][i][b][d]: contiguous in d -> two coalesced float4 stores.
      float* dst = uhat + ((size_t)(j * I_ + i) * B_ + h * 16 + n) * D_ + 8 * hi;
      *(float4*)(dst)     = make_float4(c[0], c[1], c[2], c[3]);
      *(float4*)(dst + 4) = make_float4(c[4], c[5], c[6], c[7]);
    }
  }
}

// ---------------------------------------------------------------------------
// Kernel 2: per (b, i-chunk): c = softmax_j(b_logits), partial s[j][d] += c*uhat,
// flushed with global f32 atomics (s is only 64 KB, L2-resident).
// ---------------------------------------------------------------------------
__global__ __launch_bounds__(256) void routing_softmax_s_kernel(
    const float* __restrict__ bmat, const float* __restrict__ uhat,
    float* __restrict__ s) {
  const int b  = blockIdx.y;
  const int i0 = blockIdx.x * CH;
  const int t  = threadIdx.x;

  __shared__ float c_lds[J_][CH];            // 8 KB
  for (int idx = t; idx < J_ * CH; idx += 256) {
    const int j = idx / CH, ii = idx % CH;
    c_lds[j][ii] = bmat[(size_t)(b * J_ + j) * I_ + i0 + ii];
  }
  __syncthreads();

  if (t < CH) {                              // column softmax over J
    float mx = -INFINITY;
    #pragma unroll
    for (int j = 0; j < J_; ++j) mx = fmaxf(mx, c_lds[j][t]);
    float sum = 0.f;
    #pragma unroll
    for (int j = 0; j < J_; ++j) {
      const float e = __expf(c_lds[j][t] - mx);
      c_lds[j][t] = e; sum += e;
    }
    const float inv = 1.f / sum;
    #pragma unroll
    for (int j = 0; j < J_; ++j) c_lds[j][t] *= inv;
  }
  __syncthreads();

  const int j = t >> 4;                      // 0..15 (pairs with j+16)
  const int d = t & 15;
  float acc0 = 0.f, acc1 = 0.f;
  for (int ii = 0; ii < CH; ++ii) {
    const int i = i0 + ii;
    acc0 += c_lds[j][ii]      * uhat[((size_t)( j       * I_ + i) * B_ + b) * D_ + d];
    acc1 += c_lds[j + 16][ii] * uhat[((size_t)((j + 16) * I_ + i) * B_ + b) * D_ + d];
  }
  atomicAdd(&s[(b * J_ + j) * D_ + d],      acc0);
  atomicAdd(&s[(b * J_ + j + 16) * D_ + d], acc1);
}

// ---------------------------------------------------------------------------
// Kernel 3: squash. One thread per (b,j) capsule.
// ---------------------------------------------------------------------------
__global__ __launch_bounds__(256) void squash_kernel(
    const float* __restrict__ s, float* __restrict__ v) {
  const int g = blockIdx.x * blockDim.x + threadIdx.x;
  if (g >= B_ * J_) return;
  const float* sp = s + g * D_;
  float vals[D_], n2 = 0.f;
  #pragma unroll
  for (int d = 0; d < D_; ++d) { vals[d] = sp[d]; n2 += vals[d] * vals[d]; }
  const float scale = (n2 / (1.f + n2)) * rsqrtf(n2 + 1e-7f);
  float* vp = v + g * D_;
  #pragma unroll
  for (int d = 0; d < D_; ++d) vp[d] = vals[d] * scale;
}

// ---------------------------------------------------------------------------
// Kernel 4: b[b,j,i] += dot_d(u_hat[b,j,i,:], v[b,j,:]).
// i fastest across lanes -> coalesced b updates; 64B u_hat reads per lane.
// ---------------------------------------------------------------------------
__global__ __launch_bounds__(256) void logits_update_kernel(
    const float* __restrict__ uhat, const float* __restrict__ v,
    float* __restrict__ bmat) {
  const size_t g = (size_t)blockIdx.x * blockDim.x + threadIdx.x;
  const int i = (int)(g % I_);
  const int j = (int)((g / I_) % J_);
  const int b = (int)(g / ((size_t)I_ * J_));
  const float4* up = (const float4*)(uhat + ((size_t)(j * I_ + i) * B_ + b) * D_);
  const float4* vp = (const float4*)(v + (size_t)(b * J_ + j) * D_);
  float dot = 0.f;
  #pragma unroll
  for (int q = 0; q < 4; ++q) {
    const float4 u4 = up[q], v4 = vp[q];
    dot += u4.x * v4.x + u4.y * v4.y + u4.z * v4.z + u4.w * v4.w;
  }
  bmat[(size_t)(b * J_ + j) * I_ + i] += dot;
}

// ---------------------------------------------------------------------------
extern "C" void kernel_launch(void* const* d_in, const int* in_sizes, int n_in,
                              void* d_out, int out_size, void* d_ws, size_t ws_size,
                              hipStream_t stream) {
  (void)in_sizes; (void)n_in; (void)out_size; (void)ws_size;
  const float* x = (const float*)d_in[0];   // [B, I, K]
  const float* W = (const float*)d_in[1];   // [J, I, D, K]
  float* out = (float*)d_out;               // [B, J, D]

  float* uhat = (float*)d_ws;                             // [J][I][B][D] 134 MB
  float* bmat = uhat + (size_t)J_ * I_ * B_ * D_;         // [B][J][I]    8.4 MB
  float* s    = bmat + (size_t)B_ * J_ * I_;              // [B][J][D]    64 KB
  float* v    = s + B_ * J_ * D_;                         // [B][J][D]    64 KB

  hipMemsetAsync(bmat, 0, sizeof(float) * (size_t)B_ * J_ * I_, stream);
  uhat_wmma_kernel<<<I_, 256, 0, stream>>>(x, W, uhat);

  for (int r = 0; r < 3; ++r) {
    hipMemsetAsync(s, 0, sizeof(float) * B_ * J_ * D_, stream);
    routing_softmax_s_kernel<<<dim3(I_ / CH, B_), 256, 0, stream>>>(bmat, uhat, s);
    squash_kernel<<<(B_ * J_ + 255) / 256, 256, 0, stream>>>(s, (r == 2) ? out : v);
    if (r < 2)
      logits_update_kernel<<<(B_ * J_ * I_) / 256, 256, 0, stream>>>(uhat, v, bmat);
  }
}